// SNDE_52467320488089
// MI455X (gfx1250) — compile-verified
//
#include <hip/hip_runtime.h>
#include <math.h>

// CDNA5 (gfx1250) wave32 WMMA implementation of the stabilized neural-ODE scan.
// One launch; each block time-steps a 16-row tile for all 100 Euler steps.

typedef float v2f __attribute__((ext_vector_type(2)));
typedef float v8f __attribute__((ext_vector_type(8)));

#define BSZ    4096
#define DIMN   64
#define HIDN   256
#define TSTEP  101
#define GAMMA_C 0.1f
#define EPS_C   1e-12f
#define YS 68    // padded row stride for y tile (16x64)  -> conflict-free A loads
#define HS 260   // padded row stride for h tile (16x256) -> conflict-free A loads

// gfx1250 has a hardware tanh transcendental (v_tanh_f32). Use it if the
// toolchain exposes the builtin; otherwise fall back to libm tanhf.
__device__ __forceinline__ float hw_tanh(float x) {
#if __has_builtin(__builtin_amdgcn_tanhf)
    return __builtin_amdgcn_tanhf(x);
#elif __has_builtin(__builtin_amdgcn_tanh_f32)
    return __builtin_amdgcn_tanh_f32(x);
#else
    return tanhf(x);
#endif
}

__global__ __launch_bounds__(512)
void snde_stab_ode_kernel(const float* __restrict__ y0,
                          const float* __restrict__ tarr,
                          const float* __restrict__ W1,
                          const float* __restrict__ b1,
                          const float* __restrict__ W2,
                          const float* __restrict__ b2,
                          const float* __restrict__ wg,
                          const float* __restrict__ bgp,
                          float* __restrict__ out)
{
    __shared__ float y_lds[16 * YS];          // state tile (padded)
    __shared__ float h_lds[16 * HS];          // hidden tile (padded)
    __shared__ float fp_lds[4][16 * DIMN];    // split-K partials of f (deterministic reduce)
    __shared__ float wg_s[DIMN];
    __shared__ float b2_s[DIMN];
    __shared__ float coef_s[16];
    __shared__ float wgss_s;

    const int tid  = threadIdx.x;
    const int w    = tid >> 5;      // wave id 0..15
    const int lane = tid & 31;
    const int lm   = lane & 15;     // lane % 16
    const int lh   = lane >> 4;     // lane / 16
    const int row0 = blockIdx.x * 16;

    // ---- preload W1 B-fragments: wave w owns HID columns [16w, 16w+16) ----
    // B layout (f32 4x16): element (K = r + 2*lh, N = lm) per k-step of 4.
    v2f b1f[16];
    const int n1 = w * 16 + lm;
#pragma unroll
    for (int kk = 0; kk < 16; ++kk) {
        const int k0 = kk * 4 + 2 * lh;
        b1f[kk].x = W1[(k0 + 0) * HIDN + n1];
        b1f[kk].y = W1[(k0 + 1) * HIDN + n1];
    }
    // ---- preload W2 B-fragments: wave = (kgroup g2 = w/4, ntile n2t = w%4) ----
    const int g2  = w >> 2;
    const int n2t = w & 3;
    const int n2  = n2t * 16 + lm;
    v2f b2f[16];
#pragma unroll
    for (int kk = 0; kk < 16; ++kk) {
        const int k0 = g2 * 64 + kk * 4 + 2 * lh;
        b2f[kk].x = W2[(k0 + 0) * DIMN + n2];
        b2f[kk].y = W2[(k0 + 1) * DIMN + n2];
    }
    const float b1v = b1[n1];

    if (tid < DIMN) { wg_s[tid] = wg[tid]; b2_s[tid] = b2[tid]; }
    if (tid == 0) {
        float s = 0.f;
        for (int j = 0; j < DIMN; ++j) { const float v = wg[j]; s += v * v; }
        wgss_s = s;
    }
    // stage y0 into LDS and emit state 0
    for (int e = tid; e < 16 * DIMN; e += 512) {
        const int i = e >> 6, j = e & 63;
        const float v = y0[(row0 + i) * DIMN + j];
        y_lds[i * YS + j] = v;
        out[(size_t)(row0 + i) * DIMN + j] = v;
    }
    const float dt = tarr[1] - tarr[0];
    const float bg = bgp[0];
    __syncthreads();

    for (int t = 0; t < TSTEP - 1; ++t) {
        // ---- phase A: this wave's 16x16 tile of h = tanh(y @ W1 + b1) ----
        v8f acc1 = {0.f, 0.f, 0.f, 0.f, 0.f, 0.f, 0.f, 0.f};
#pragma unroll
        for (int kk = 0; kk < 16; ++kk) {
            // A layout (f32 16x4): element (M = lm, K = r + 2*lh); 8B-aligned pair load
            const v2f a = *(const v2f*)&y_lds[lm * YS + kk * 4 + 2 * lh];
            acc1 = __builtin_amdgcn_wmma_f32_16x16x4_f32(
                       false, a, false, b1f[kk], (short)0, acc1, false, false);
        }
        // C layout: (M = r + 8*lh, N = lm) -> activation + spill to LDS
#pragma unroll
        for (int r = 0; r < 8; ++r) {
            h_lds[(r + 8 * lh) * HS + w * 16 + lm] = hw_tanh(acc1[r] + b1v);
        }
        // constraint coefficient per row (wave 15; all 32 lanes split the dot)
        if (w == 15) {
            float p = 0.f;
            const int j0 = lh * 32;
#pragma unroll
            for (int j = 0; j < 32; ++j) p += y_lds[lm * YS + j0 + j] * wg_s[j0 + j];
            const float z = bg + p + __shfl_xor(p, 16, 32);   // combine the two halves
            const float gv = hw_tanh(z);
            const float c  = 1.f - gv * gv;
            const float s  = c * c * wgss_s;                  // G G^T (1x1)
            const float pinv = (fabsf(s) > EPS_C) ? (1.f / s) : 0.f;
            if (lh == 0) coef_s[lm] = GAMMA_C * gv * pinv * c; // stable = coef * wg
        }
        __syncthreads();

        // ---- phase B: split-K partial of f = h @ W2 over this wave's 64-wide K slice ----
        v8f acc2 = {0.f, 0.f, 0.f, 0.f, 0.f, 0.f, 0.f, 0.f};
#pragma unroll
        for (int kk = 0; kk < 16; ++kk) {
            const v2f a = *(const v2f*)&h_lds[lm * HS + g2 * 64 + kk * 4 + 2 * lh];
            acc2 = __builtin_amdgcn_wmma_f32_16x16x4_f32(
                       false, a, false, b2f[kk], (short)0, acc2, false, false);
        }
#pragma unroll
        for (int r = 0; r < 8; ++r) {
            fp_lds[g2][(r + 8 * lh) * DIMN + n2] = acc2[r];
        }
        __syncthreads();

        // ---- phase C: fixed-order K reduction + Euler update + stream output ----
        const size_t obase = (size_t)(t + 1) * BSZ * DIMN;
        for (int e = tid; e < 16 * DIMN; e += 512) {
            const int i = e >> 6, j = e & 63;
            const float fv = b2_s[j] + fp_lds[0][e] + fp_lds[1][e]
                                     + fp_lds[2][e] + fp_lds[3][e];
            const float yn = y_lds[i * YS + j] + (fv - coef_s[i] * wg_s[j]) * dt;
            y_lds[i * YS + j] = yn;
            out[obase + (size_t)(row0 + i) * DIMN + j] = yn;
        }
        __syncthreads();
    }
}

extern "C" void kernel_launch(void* const* d_in, const int* in_sizes, int n_in,
                              void* d_out, int out_size, void* d_ws, size_t ws_size,
                              hipStream_t stream) {
    (void)in_sizes; (void)n_in; (void)out_size; (void)d_ws; (void)ws_size;
    const float* y0 = (const float*)d_in[0];
    const float* t  = (const float*)d_in[1];
    const float* W1 = (const float*)d_in[2];
    const float* b1 = (const float*)d_in[3];
    const float* W2 = (const float*)d_in[4];
    const float* b2 = (const float*)d_in[5];
    const float* wg = (const float*)d_in[6];
    const float* bg = (const float*)d_in[7];
    float* out = (float*)d_out;

    dim3 grid(BSZ / 16);   // 256 blocks, one 16-row tile each
    dim3 block(512);       // 16 wave32s per block
    snde_stab_ode_kernel<<<grid, block, 0, stream>>>(y0, t, W1, b1, W2, b2, wg, bg, out);
}